// GCNConv_dgl_3513283248909
// MI455X (gfx1250) — compile-verified
//
#include <hip/hip_runtime.h>
#include <hip/hip_bf16.h>

typedef float v2f __attribute__((ext_vector_type(2)));
typedef float v8f __attribute__((ext_vector_type(8)));

#define D 64  // D_IN == D_OUT == 64

// ---------------------------------------------------------------------------
// Kernel 1: h = x @ W^T + b   via V_WMMA_F32_16X16X4_F32
// One wave -> one 16x16 tile of h. Block = 4 waves covering all 64 out cols.
// A (16x4 f32): lane l holds M=l%16, K=(l/16)*2+{0,1}  -> float2 load
// B (4x16 f32): lane l holds N=l%16, K=(l/16)*2+{0,1}  -> float2 load of W row
// C/D (16x16 f32): VGPR r, lanes 0-15 -> M=r, lanes 16-31 -> M=r+8, N=lane%16
// ---------------------------------------------------------------------------
__global__ __launch_bounds__(128)
void gcn_gemm_bias(const float* __restrict__ x,   // [N, 64]
                   const float* __restrict__ W,   // [64, 64] (row = out feat)
                   const float* __restrict__ b,   // [64]
                   float* __restrict__ h,         // [N, 64]
                   int n_nodes) {
  const int lane = threadIdx.x & 31;
  const int wave = threadIdx.x >> 5;          // 0..3 -> which 16-col chunk
  const int row0 = blockIdx.x << 4;           // 16 rows per block
  const int col0 = wave << 4;

  const int m  = lane & 15;                   // A row / B col within tile
  const int kk = (lane >> 4) << 1;            // K sub-offset: 0 or 2

  // Clamp load row for (non-existent here) tail tiles so EXEC stays all-1s.
  int arow = row0 + m;
  if (arow >= n_nodes) arow = n_nodes - 1;

  const float* __restrict__ xrow = x + (size_t)arow * D;
  const float* __restrict__ wrow = W + (size_t)(col0 + m) * D;

  v8f acc = {};
#pragma unroll
  for (int k0 = 0; k0 < D; k0 += 4) {
    v2f a  = *(const v2f*)(xrow + k0 + kk);
    v2f bb = *(const v2f*)(wrow + k0 + kk);
    // (neg_a, A, neg_b, B, c_mod, C, reuse_a, reuse_b)
    acc = __builtin_amdgcn_wmma_f32_16x16x4_f32(
        false, a, false, bb, (short)0, acc, false, false);
  }

  const float bias = b[col0 + m];
  const int rbase = row0 + ((lane >> 4) << 3);   // upper half-wave -> rows +8
#pragma unroll
  for (int r = 0; r < 8; ++r) {
    const int row = rbase + r;
    if (row < n_nodes) {
      h[(size_t)row * D + col0 + m] = acc[r] + bias;
    }
  }
}

// ---------------------------------------------------------------------------
// Kernel 2: edge scatter-add. 16 lanes per edge, float4 gather (256B/edge,
// one full cacheline), 4x global_atomic_add_f32 into out[dst]. h and out both
// fit in the 192MB L2, so this is L2-resident.
// ---------------------------------------------------------------------------
__global__ __launch_bounds__(256)
void gcn_edge_scatter(const float* __restrict__ h,
                      const int* __restrict__ src,
                      const int* __restrict__ dst,
                      const float* __restrict__ w,
                      float* __restrict__ out,
                      int n_edges) {
  const int gid = blockIdx.x * 256 + threadIdx.x;
  const int e = gid >> 4;
  if (e >= n_edges) return;
  const int f = (gid & 15) << 2;     // feature offset: 0,4,...,60

  const int   s  = src[e];
  const int   d  = dst[e];
  const float wv = w[e];

  const float4 hv = *(const float4*)(h + (size_t)s * D + f);
  float* __restrict__ o = out + (size_t)d * D + f;

  atomicAdd(o + 0, hv.x * wv);
  atomicAdd(o + 1, hv.y * wv);
  atomicAdd(o + 2, hv.z * wv);
  atomicAdd(o + 3, hv.w * wv);
}

// ---------------------------------------------------------------------------
// Launch: memset(out) -> GEMM+bias into d_ws -> edge scatter-add.
// ---------------------------------------------------------------------------
extern "C" void kernel_launch(void* const* d_in, const int* in_sizes, int n_in,
                              void* d_out, int out_size, void* d_ws, size_t ws_size,
                              hipStream_t stream) {
  const float* x   = (const float*)d_in[0];
  const int*   src = (const int*)  d_in[1];
  const int*   dst = (const int*)  d_in[2];
  const float* w   = (const float*)d_in[3];
  const float* W   = (const float*)d_in[4];
  const float* b   = (const float*)d_in[5];
  float*       out = (float*)d_out;
  float*       h   = (float*)d_ws;             // [n_nodes, 64] = 25.6 MB

  const int n_nodes = in_sizes[0] / D;
  const int n_edges = in_sizes[1];

  // Harness poisons d_out; zero it (async memset is graph-capture safe).
  hipMemsetAsync(d_out, 0, (size_t)out_size * sizeof(float), stream);

  const int row_tiles = (n_nodes + 15) / 16;   // 6250
  gcn_gemm_bias<<<dim3(row_tiles), dim3(128), 0, stream>>>(x, W, b, h, n_nodes);

  const long long threads = (long long)n_edges * 16;
  const int blocks = (int)((threads + 255) / 256);  // 75000
  gcn_edge_scatter<<<dim3(blocks), dim3(256), 0, stream>>>(h, src, dst, w, out, n_edges);
}